// CenterNetLoss_45621142618500
// MI455X (gfx1250) — compile-verified
//
#include <hip/hip_runtime.h>
#include <hip/hip_bf16.h>
#include <stdint.h>

// Problem constants (match reference)
#define Bq    16
#define Cq    80
#define Hq    128
#define Wq    128
#define Nq    50
#define CANDq 100

#define NBLK 2048   // focal-baseline grid
#define NTHR 256

typedef float v2f __attribute__((ext_vector_type(2)));
typedef float v8f __attribute__((ext_vector_type(8)));

// Wave32 sum via V_WMMA_F32_16X16X4_F32:
// A (16x4 f32): VGPR0 = per-lane value (K=0 for lanes0-15, K=2 for lanes16-31),
// VGPR1 = 0.  B = all ones (layout-agnostic).  D[m,n] = s_m + s_{m+16}.
// Thread-local sum of the 8 D regs = half-wave sum; xor-shuffle by 16 finishes.
__device__ __forceinline__ float wave_sum_wmma(float v) {
  v2f a; a[0] = v;    a[1] = 0.0f;
  v2f bm; bm[0] = 1.0f; bm[1] = 1.0f;
  v8f c = {0.f,0.f,0.f,0.f,0.f,0.f,0.f,0.f};
  v8f d = __builtin_amdgcn_wmma_f32_16x16x4_f32(false, a, false, bm,
                                                (short)0, c, false, false);
  float t = ((d[0]+d[1])+(d[2]+d[3])) + ((d[4]+d[5])+(d[6]+d[7]));
  t += __shfl_xor(t, 16, 32);
  return t;
}

// Block sum; result valid on thread 0. Must be called by ALL threads
// (uniform control flow -> EXEC all ones for the WMMA).
__device__ __forceinline__ float block_sum(float v, float* sw) {
  float t = wave_sum_wmma(v);
  int lane = threadIdx.x & 31;
  int wid  = threadIdx.x >> 5;
  int nw   = blockDim.x >> 5;
  if (lane == 0) sw[wid] = t;
  __syncthreads();
  float s = 0.0f;
  if (threadIdx.x == 0) for (int w = 0; w < nw; ++w) s += sw[w];
  __syncthreads();
  return s;
}

// focal negative term at gt==0: p^2 * log(1-p), p clipped to [1e-4, 0.9999]
__device__ __forceinline__ float neg0_term(float x) {
  float p = fminf(fmaxf(x, 1e-4f), 0.9999f);
  return p * p * logf(1.0f - p);
}

// ---------------------------------------------------------------------------
// Kernel 1: bandwidth-bound streaming pass over cls_pred (84 MB), gt=0 baseline.
// 2048 blocks x 256 thr x 10 float4 iterations == 20,971,520 elements exactly.
// ---------------------------------------------------------------------------
__global__ void k_focal_base(const float* __restrict__ cls,
                             float* __restrict__ partials) {
  const int M4 = Bq*Cq*Hq*Wq/4;  // 5,242,880
  const float4* p4 = (const float4*)cls;
  int stride = gridDim.x * blockDim.x;
  float acc = 0.0f;
  for (int i = blockIdx.x*blockDim.x + threadIdx.x; i < M4; i += stride) {
    float4 x = p4[i];
    acc += neg0_term(x.x) + neg0_term(x.y) + neg0_term(x.z) + neg0_term(x.w);
  }
  __shared__ float sw[NTHR/32];
  float s = block_sum(acc, sw);
  if (threadIdx.x == 0) partials[blockIdx.x] = s;
}

// ---------------------------------------------------------------------------
// Kernel 2: sparse heatmap correction. One block per batch, <=450 candidates,
// scatter-max dedup (winner = max gt, lowest index), patch focal term.
// ---------------------------------------------------------------------------
__global__ void k_heat_corr(const float* __restrict__ cls,
                            const float* __restrict__ gt_box,
                            const int*   __restrict__ gt_class,
                            float* __restrict__ corr) {
  const int b = blockIdx.x;
  const int tid = threadIdx.x;
  const int NC = Nq*9;
  __shared__ int   scx[Nq], scy[Nq], sch[Nq];
  __shared__ unsigned char sval[Nq], sintr[Nq];
  __shared__ int   cpos[NC];
  __shared__ float cval[NC];

  if (tid < Nq) {
    int gc = gt_class[b*Nq + tid];
    bool valid = (gc != -1);
    const float* bx = gt_box + ((size_t)b*Nq + tid)*4;
    float wv = bx[2]-bx[0], hv = bx[3]-bx[1];
    int cx = (int)(floorf(floorf(wv/2.0f)/4.0f));
    int cy = (int)(floorf(floorf(hv/2.0f)/4.0f));
    scx[tid]=cx; scy[tid]=cy; sch[tid]=max(gc,0);
    sval[tid]  = valid ? 1 : 0;
    sintr[tid] = (valid && cx>=1 && cy>=1 && cx+1<Hq && cy+1<Wq) ? 1 : 0;
  }
  __syncthreads();

  const int dxs[9] = {0,-1,-1,-1, 0, 0, 1, 1, 1};
  const int dys[9] = {0,-1, 0, 1,-1, 1,-1, 0, 1};
  const float ONE_V = (float)0.6065306597126334;   // exp(-0.5)
  const float TWO_V = (float)0.36787944117144233;  // exp(-1.0)
  const float vvs[9] = {1.0f, TWO_V, ONE_V, TWO_V, ONE_V, ONE_V, TWO_V, ONE_V, TWO_V};

  if (tid < NC) {
    int n = tid/9, d = tid%9;
    int x = scx[n]+dxs[d]; x = min(max(x,0),Hq-1);
    int y = scy[n]+dys[d]; y = min(max(y,0),Wq-1);
    float v = (d==0) ? (sval[n] ? 1.0f : 0.0f) : (sintr[n] ? vvs[d] : 0.0f);
    cpos[tid] = (sch[n]*Hq + x)*Wq + y;
    cval[tid] = v;
  }
  __syncthreads();

  float delta = 0.0f;
  if (tid < NC && cval[tid] > 0.0f) {
    int pos = cpos[tid];
    float g = cval[tid];
    bool win = true;
    for (int j = 0; j < NC; ++j) {
      if (j == tid) continue;
      if (cpos[j] == pos) {
        float gj = cval[j];
        if (gj > g || (gj == g && j < tid)) { win = false; break; }
      }
    }
    if (win) {
      float raw = cls[(size_t)b*Cq*Hq*Wq + pos];
      float p = fminf(fmaxf(raw, 1e-4f), 0.9999f);
      float base = p*p*logf(1.0f - p);           // already counted in kernel 1
      if (g == 1.0f) {
        float q = 1.0f - p; float q4 = (q*q)*(q*q);
        delta = q4*logf(p) - base;               // pos term replaces neg term
      } else {
        float q = 1.0f - g; float q4 = (q*q)*(q*q);
        delta = (q4 - 1.0f)*base;                // rescale neg term by (1-g)^4
      }
    }
  }
  __shared__ float sw[512/32];
  float s = block_sum(delta, sw);
  if (tid == 0) corr[b] = s;
}

// ---------------------------------------------------------------------------
// Kernel 3: per-batch top-CAND smallest cls values in the box window.
// Keys pack (order-preserving float bits << 32 | flat index) so min-key order
// == top_k(-vals) order incl. tie-break. Owner thread accumulates L1 losses.
// ---------------------------------------------------------------------------
__global__ void k_target(const float* __restrict__ cls,
                         const float* __restrict__ offp,
                         const float* __restrict__ sizep,
                         const float* __restrict__ gt_box,
                         const int*   __restrict__ gt_class,
                         float* __restrict__ cntW,
                         float* __restrict__ offW,
                         float* __restrict__ sizeW) {
  const int b = blockIdx.x;
  const int tid = threadIdx.x;
  __shared__ unsigned long long keys[64*64];   // window <= 64x64 cells
  __shared__ unsigned long long red[256];
  __shared__ int s_ch, s_left, s_top, s_cols, s_cells, s_K;
  __shared__ float s_wv, s_hv, s_o0, s_o1;

  if (tid == 0) {
    int last = 0; bool anyv = false;
    for (int n = 0; n < Nq; ++n)
      if (gt_class[b*Nq+n] != -1) { last = n; anyv = true; }
    const float* bx = gt_box + ((size_t)b*Nq + last)*4;
    float wv = bx[2]-bx[0], hv = bx[3]-bx[1];
    int ch = max(gt_class[b*Nq+last], 0);
    int cx = (int)(floorf(floorf(wv/2.0f)/4.0f));
    int cy = (int)(floorf(floorf(hv/2.0f)/4.0f));
    int w4 = (int)floorf(wv/4.0f);
    int h4 = (int)floorf(hv/4.0f);
    auto fd = [](int a, int d) { int q = a/d; if ((a%d) && ((a<0)!=(d<0))) --q; return q; };
    int left   = max(fd(cx - fd(w4,2), 2), 0);
    int right  = min(fd(cx + fd(w4,2), 2), Hq/2);
    int top    = max(fd(cy - fd(h4,2), 2), 0);
    int bottom = min(fd(cy + fd(h4,2), 2), Wq/2);
    int rows = right-left, cols = bottom-top;
    int cells = (anyv && rows > 0 && cols > 0) ? rows*cols : 0;
    s_ch = ch; s_left = left; s_top = top;
    s_cols = max(cols, 1); s_cells = cells;
    s_K = min(CANDq, cells);
    s_wv = wv; s_hv = hv;
    float cxf = wv/8.0f, cyf = hv/8.0f;      // (wv/2)/4
    s_o0 = cxf - floorf(cxf);
    s_o1 = cyf - floorf(cyf);
  }
  __syncthreads();

  const int cells = s_cells, cols = s_cols, left = s_left, top = s_top, ch = s_ch;
  for (int idx = tid; idx < cells; idx += blockDim.x) {
    int r = idx/cols, c2 = idx%cols;
    int i = left + r, j = top + c2;
    float v = cls[((size_t)b*Cq + ch)*(size_t)(Hq*Wq) + (size_t)i*Wq + j];
    unsigned ub = __float_as_uint(v);
    ub = (ub & 0x80000000u) ? ~ub : (ub | 0x80000000u);  // monotone map
    keys[idx] = ((unsigned long long)ub << 32) | (unsigned)(i*Wq + j);
  }
  __syncthreads();

  // per-thread cached local min over owned (strided) cells
  unsigned long long lmKey = ~0ull; int lmIdx = -1;
  for (int idx = tid; idx < cells; idx += blockDim.x)
    if (keys[idx] < lmKey) { lmKey = keys[idx]; lmIdx = idx; }

  float cntT = 0.f, offT = 0.f, sizeT = 0.f;
  const int K = s_K;
  for (int k = 0; k < K; ++k) {
    red[tid] = lmKey;
    __syncthreads();
    for (int s = blockDim.x >> 1; s > 0; s >>= 1) {
      if (tid < s) { unsigned long long o = red[tid+s]; if (o < red[tid]) red[tid] = o; }
      __syncthreads();
    }
    unsigned long long mk = red[0];
    __syncthreads();
    if (lmKey == mk) {            // unique owner (keys contain distinct indices)
      unsigned flat = (unsigned)(mk & 0xFFFFFFFFull);
      int i = (int)(flat / Wq), j = (int)(flat % Wq);
      float o0 = offp [((size_t)b*2 + 0)*(Hq*Wq) + (size_t)i*Wq + j];
      float o1 = offp [((size_t)b*2 + 1)*(Hq*Wq) + (size_t)i*Wq + j];
      float z0 = sizep[((size_t)b*2 + 0)*(Hq*Wq) + (size_t)i*Wq + j];
      float z1 = sizep[((size_t)b*2 + 1)*(Hq*Wq) + (size_t)i*Wq + j];
      offT  += fabsf(o0 - s_o0) + fabsf(o1 - s_o1);
      sizeT += fabsf(z0 - s_wv) + fabsf(z1 - s_hv);
      cntT  += 1.0f;
      keys[lmIdx] = ~0ull;        // remove and refresh cached local min
      lmKey = ~0ull; lmIdx = -1;
      for (int idx = tid; idx < cells; idx += blockDim.x)
        if (keys[idx] < lmKey) { lmKey = keys[idx]; lmIdx = idx; }
    }
    __syncthreads();
  }

  __shared__ float sw[256/32];
  float c  = block_sum(cntT,  sw);
  float of = block_sum(offT,  sw);
  float sz = block_sum(sizeT, sw);
  if (tid == 0) { cntW[b] = c; offW[b] = of; sizeW[b] = sz; }
}

// ---------------------------------------------------------------------------
// Kernel 4: deterministic final combine -> scalar loss.
// ---------------------------------------------------------------------------
__global__ void k_final(const float* __restrict__ partials,
                        const float* __restrict__ corr,
                        const float* __restrict__ cntW,
                        const float* __restrict__ offW,
                        const float* __restrict__ sizeW,
                        float* __restrict__ out) {
  int tid = threadIdx.x;
  float v = 0.f;
  for (int i = tid; i < NBLK; i += blockDim.x) v += partials[i];
  if (tid < Bq) v += corr[tid];
  __shared__ float sw[256/32];
  float S = block_sum(v, sw);          // total focal sum (pos+neg)
  if (tid == 0) {
    float np = 0.f, off = 0.f, sz = 0.f;
    for (int b = 0; b < Bq; ++b) { np += cntW[b]; off += offW[b]; sz += sizeW[b]; }
    float npos = fmaxf(np, 1.0f);
    float cls_loss = -S / (float)(Bq*Hq*Wq);
    out[0] = cls_loss + off/npos + 0.1f*sz/npos;
  }
}

extern "C" void kernel_launch(void* const* d_in, const int* in_sizes, int n_in,
                              void* d_out, int out_size, void* d_ws, size_t ws_size,
                              hipStream_t stream) {
  const float* cls  = (const float*)d_in[0];
  const float* offp = (const float*)d_in[1];
  const float* szp  = (const float*)d_in[2];
  const float* gbox = (const float*)d_in[3];
  const int*   gcls = (const int*)  d_in[4];
  float* ws = (float*)d_ws;
  float* partials = ws;               // [NBLK]
  float* corr  = partials + NBLK;     // [Bq]
  float* cntW  = corr  + Bq;          // [Bq]
  float* offW  = cntW  + Bq;          // [Bq]
  float* sizeW = offW  + Bq;          // [Bq]  total: (2048+64)*4 = 8448 bytes

  k_focal_base<<<NBLK, NTHR, 0, stream>>>(cls, partials);
  k_heat_corr <<<Bq,   512,  0, stream>>>(cls, gbox, gcls, corr);
  k_target    <<<Bq,   256,  0, stream>>>(cls, offp, szp, gbox, gcls, cntW, offW, sizeW);
  k_final     <<<1,    256,  0, stream>>>(partials, corr, cntW, offW, sizeW, (float*)d_out);

  (void)in_sizes; (void)n_in; (void)out_size; (void)ws_size;
}